// JetGraphAutoencoderClassification_56092272886264
// MI455X (gfx1250) — compile-verified
//
#include <hip/hip_runtime.h>
#include <hip/hip_bf16.h>
#include <math.h>

typedef __attribute__((ext_vector_type(2))) float v2f;
typedef __attribute__((ext_vector_type(8))) float v8f;

// ---------------- order-preserving float<->uint encoding for atomic max ----
__device__ __forceinline__ unsigned encf(float f) {
    unsigned u = __float_as_uint(f);
    return (u & 0x80000000u) ? ~u : (u | 0x80000000u);
}
__device__ __forceinline__ float decf(unsigned e) {
    unsigned u = (e & 0x80000000u) ? (e & 0x7fffffffu) : ~e;
    return __uint_as_float(u);
}
#define ENC_NEG_INF 0x007FFFFFu   // encf(-inf)

// ---------------- init aggregation buffer to enc(-inf) --------------------
__global__ void __launch_bounds__(256) init_enc_kernel(unsigned* __restrict__ p, int n) {
    int i = blockIdx.x * 256 + threadIdx.x;
    if (i < n) p[i] = ENC_NEG_INF;
}

// ---------------- EdgeConv over explicit edge list (atomic max) ------------
// message = MLP2([x_i, x_j - x_i]) ; segment_max by dst via atomicMax(enc)
template<int F, int FH, int FO>
__global__ void __launch_bounds__(256) edge_conv_edges(
    const float* __restrict__ xin, const int* __restrict__ src,
    const int* __restrict__ dst,
    const float* __restrict__ w1, const float* __restrict__ b1,
    const float* __restrict__ w2, const float* __restrict__ b2,
    unsigned* __restrict__ agg, int E)
{
    constexpr int FI = 2 * F;
    __shared__ float sw1[FI * FH], sb1[FH], sw2[FH * FO], sb2[FO];
    for (int i = threadIdx.x; i < FI * FH; i += 256) sw1[i] = w1[i];
    for (int i = threadIdx.x; i < FH * FO; i += 256) sw2[i] = w2[i];
    if (threadIdx.x < FH) sb1[threadIdx.x] = b1[threadIdx.x];
    if (threadIdx.x < FO) sb2[threadIdx.x] = b2[threadIdx.x];
    __syncthreads();

    int e = blockIdx.x * 256 + threadIdx.x;
    if (e >= E) return;
    int s = src[e], d = dst[e];
    float in[FI];
    const float* pi = xin + (size_t)d * F;
    const float* pj = xin + (size_t)s * F;
#pragma unroll
    for (int f = 0; f < F; ++f) { float a = pi[f]; in[f] = a; in[F + f] = pj[f] - a; }
    float hid[FH];
#pragma unroll
    for (int j = 0; j < FH; ++j) {
        float a = sb1[j];
#pragma unroll
        for (int i = 0; i < FI; ++i) a = fmaf(in[i], sw1[i * FH + j], a);
        hid[j] = fmaxf(a, 0.0f);
    }
    unsigned* base = agg + (size_t)d * FO;
#pragma unroll
    for (int j = 0; j < FO; ++j) {
        float a = sb2[j];
#pragma unroll
        for (int k = 0; k < FH; ++k) a = fmaf(hid[k], sw2[k * FO + j], a);
        atomicMax(base + j, encf(a));
    }
}

// ---------------- decode agg -> relu(where(isfinite, agg, 0)) --------------
__global__ void __launch_bounds__(256) decode_relu_kernel(
    const unsigned* __restrict__ a, float* __restrict__ o, int n)
{
    int i = blockIdx.x * 256 + threadIdx.x;
    if (i < n) o[i] = fmaxf(decf(a[i]), 0.0f);   // max(-inf,0)=0 folds the isfinite rule
}

__global__ void __launch_bounds__(256) decode_h2_sq_kernel(
    const unsigned* __restrict__ a, float* __restrict__ h2,
    float* __restrict__ sq, int Nn)
{
    int nd = blockIdx.x * 256 + threadIdx.x;
    if (nd >= Nn) return;
    float s = 0.0f;
#pragma unroll
    for (int j = 0; j < 8; ++j) {
        float v = fmaxf(decf(a[nd * 8 + j]), 0.0f);
        h2[nd * 8 + j] = v;
        s = fmaf(v, v, s);
    }
    sq[nd] = s;
}

// ---------------- fused kNN: fp32 WMMA Gram + register top-8 ---------------
// One wave per 16 query rows. Gram 16x16 tile = two V_WMMA_F32_16X16X4_F32.
// A layout (ISA 7.12.2, 32-bit A 16x4): lanes 0-15 VGPR0=K0,VGPR1=K1;
// lanes 16-31 VGPR0=K2,VGPR1=K3. B (4x16) mirrored; since B[k][n]=h2[j0+n][k],
// both fragments use the same per-lane load pattern.
// C layout: VGPR v, lanes 0-15 -> (M=v, N=lane); lanes 16-31 -> (M=v+8, N=lane-16).
__global__ void __launch_bounds__(256) knn_wmma_kernel(
    const float* __restrict__ h2, const float* __restrict__ sq,
    int* __restrict__ knn, int n)
{
    __shared__ float tile[8][256];   // per-wave 16x16 distance tile
    __shared__ float md[8][128];     // per-wave merge buffer (dist)
    __shared__ int   mi[8][128];     // per-wave merge buffer (idx)

    const int lane  = threadIdx.x & 31;
    const int wave  = threadIdx.x >> 5;
    const int i0    = (blockIdx.x * 8 + wave) * 16;
    const int lrow  = lane & 15;
    const int half  = lane >> 4;      // 0 or 1
    const int khalf = half * 2;       // K offset within a K=4 fragment
    const int mbase = half * 8;       // C-row base this lane holds
    const int cbase = half * 8;       // column range this lane scans for top-k

    v2f a_lo, a_hi;
    {
        const float* p = h2 + (size_t)(i0 + lrow) * 8 + khalf;
        a_lo[0] = p[0]; a_lo[1] = p[1];
        a_hi[0] = p[4]; a_hi[1] = p[5];
    }
    float rowsq[8];
#pragma unroll
    for (int v = 0; v < 8; ++v) rowsq[v] = sq[i0 + mbase + v];

    float bd[8]; int bi[8];
#pragma unroll
    for (int s = 0; s < 8; ++s) { bd[s] = __builtin_inff(); bi[s] = 0; }
    float worst = __builtin_inff(); int wslot = 0;

    for (int j0 = 0; j0 < n; j0 += 16) {
        const float* p = h2 + (size_t)(j0 + lrow) * 8 + khalf;
        v2f b_lo, b_hi;
        b_lo[0] = p[0]; b_lo[1] = p[1];
        b_hi[0] = p[4]; b_hi[1] = p[5];
        float csq = sq[j0 + lrow];

        v8f acc = {0.f, 0.f, 0.f, 0.f, 0.f, 0.f, 0.f, 0.f};
        acc = __builtin_amdgcn_wmma_f32_16x16x4_f32(false, a_lo, false, b_lo,
                                                    (short)0, acc, false, false);
        acc = __builtin_amdgcn_wmma_f32_16x16x4_f32(false, a_hi, false, b_hi,
                                                    (short)0, acc, false, false);
#pragma unroll
        for (int v = 0; v < 8; ++v) {
            float dd = fmaf(-2.0f, acc[v], rowsq[v] + csq);
            if (j0 + lrow == i0 + mbase + v) dd = __builtin_inff();  // exclude self
            tile[wave][(mbase + v) * 16 + lrow] = dd;
        }
        __builtin_amdgcn_wave_barrier();   // DS ops are in-order within a wave
        // lane handles row `lrow`, columns cbase..cbase+7 of this tile
#pragma unroll
        for (int u = 0; u < 8; ++u) {
            float dd = tile[wave][lrow * 16 + cbase + u];
            if (dd < worst) {
                bd[wslot] = dd; bi[wslot] = j0 + cbase + u;
                worst = bd[0]; wslot = 0;
#pragma unroll
                for (int s = 1; s < 8; ++s)
                    if (bd[s] > worst) { worst = bd[s]; wslot = s; }
            }
        }
        __builtin_amdgcn_wave_barrier();
    }

    // merge the two half-row candidate lists (lanes 16-31 publish via LDS)
    if (half) {
#pragma unroll
        for (int s = 0; s < 8; ++s) {
            md[wave][lrow * 8 + s] = bd[s];
            mi[wave][lrow * 8 + s] = bi[s];
        }
    }
    __builtin_amdgcn_wave_barrier();
    if (!half) {
#pragma unroll
        for (int s = 0; s < 8; ++s) {
            float dd = md[wave][lrow * 8 + s];
            int   jj = mi[wave][lrow * 8 + s];
            if (dd < worst) {
                bd[wslot] = dd; bi[wslot] = jj;
                worst = bd[0]; wslot = 0;
#pragma unroll
                for (int t = 1; t < 8; ++t)
                    if (bd[t] > worst) { worst = bd[t]; wslot = t; }
            }
        }
        int* o = knn + (size_t)(i0 + lrow) * 8;
#pragma unroll
        for (int s = 0; s < 8; ++s) o[s] = bi[s];
    }
}

// ---------------- EdgeConv over kNN edges (8 fixed in-edges per node) ------
template<int F, int FH, int FO>
__global__ void __launch_bounds__(256) edge_conv_knn(
    const float* __restrict__ xin, const int* __restrict__ knn,
    const float* __restrict__ w1, const float* __restrict__ b1,
    const float* __restrict__ w2, const float* __restrict__ b2,
    float* __restrict__ out, int Nn)
{
    constexpr int FI = 2 * F;
    __shared__ float sw1[FI * FH], sb1[FH], sw2[FH * FO], sb2[FO];
    for (int i = threadIdx.x; i < FI * FH; i += 256) sw1[i] = w1[i];
    for (int i = threadIdx.x; i < FH * FO; i += 256) sw2[i] = w2[i];
    if (threadIdx.x < FH) sb1[threadIdx.x] = b1[threadIdx.x];
    if (threadIdx.x < FO) sb2[threadIdx.x] = b2[threadIdx.x];
    __syncthreads();

    int nd = blockIdx.x * 256 + threadIdx.x;
    if (nd >= Nn) return;
    float xi[F];
#pragma unroll
    for (int f = 0; f < F; ++f) xi[f] = xin[(size_t)nd * F + f];
    float accv[FO];
#pragma unroll
    for (int j = 0; j < FO; ++j) accv[j] = -__builtin_inff();

    for (int e = 0; e < 8; ++e) {
        int s = knn[(size_t)nd * 8 + e];
        float in[FI];
        const float* pj = xin + (size_t)s * F;
#pragma unroll
        for (int f = 0; f < F; ++f) { in[f] = xi[f]; in[F + f] = pj[f] - xi[f]; }
        float hid[FH];
#pragma unroll
        for (int j = 0; j < FH; ++j) {
            float a = sb1[j];
#pragma unroll
            for (int i = 0; i < FI; ++i) a = fmaf(in[i], sw1[i * FH + j], a);
            hid[j] = fmaxf(a, 0.0f);
        }
#pragma unroll
        for (int j = 0; j < FO; ++j) {
            float a = sb2[j];
#pragma unroll
            for (int k = 0; k < FH; ++k) a = fmaf(hid[k], sw2[k * FO + j], a);
            accv[j] = fmaxf(accv[j], a);
        }
    }
#pragma unroll
    for (int j = 0; j < FO; ++j)
        out[(size_t)nd * FO + j] = fmaxf(accv[j], 0.0f);   // relu (always finite)
}

// ---------------- per-graph max-pool + MLP head ----------------------------
__global__ void __launch_bounds__(128) head_kernel(
    const float* __restrict__ h4,
    const float* __restrict__ fc3w, const float* __restrict__ fc3b,
    const float* __restrict__ outw, const float* __restrict__ outb,
    float* __restrict__ out)
{
    __shared__ float sd[128][16];
    const int t = threadIdx.x, g = blockIdx.x;
    const float* p = h4 + ((size_t)g * 128 + t) * 16;
#pragma unroll
    for (int f = 0; f < 16; ++f) sd[t][f] = p[f];
    __syncthreads();
    for (int s = 64; s > 0; s >>= 1) {
        if (t < s) {
#pragma unroll
            for (int f = 0; f < 16; ++f) sd[t][f] = fmaxf(sd[t][f], sd[t + s][f]);
        }
        __syncthreads();
    }
    if (t == 0) {
        float z[6];
#pragma unroll
        for (int j = 0; j < 6; ++j) {
            float a = fc3b[j];
#pragma unroll
            for (int i = 0; i < 16; ++i) a = fmaf(sd[0][i], fc3w[i * 6 + j], a);
            z[j] = fmaxf(a, 0.0f);
        }
        float o = outb[0];
#pragma unroll
        for (int j = 0; j < 6; ++j) o = fmaf(z[j], outw[j], o);
        out[g] = 1.0f / (1.0f + expf(-o));
    }
}

// ---------------------------------------------------------------------------
extern "C" void kernel_launch(void* const* d_in, const int* in_sizes, int n_in,
                              void* d_out, int out_size, void* d_ws, size_t ws_size,
                              hipStream_t stream)
{
    constexpr int N = 16384, G = 128, E0 = 262144;
    const float* x    = (const float*)d_in[0];
    const int*   ei   = (const int*)d_in[1];    // [2,E0]: src row0, dst row1
    const float* c1w1 = (const float*)d_in[3];
    const float* c1b1 = (const float*)d_in[4];
    const float* c1w2 = (const float*)d_in[5];
    const float* c1b2 = (const float*)d_in[6];
    const float* c2w1 = (const float*)d_in[7];
    const float* c2b1 = (const float*)d_in[8];
    const float* c2w2 = (const float*)d_in[9];
    const float* c2b2 = (const float*)d_in[10];
    const float* c3w1 = (const float*)d_in[11];
    const float* c3b1 = (const float*)d_in[12];
    const float* c3w2 = (const float*)d_in[13];
    const float* c3b2 = (const float*)d_in[14];
    const float* c4w1 = (const float*)d_in[15];
    const float* c4b1 = (const float*)d_in[16];
    const float* c4w2 = (const float*)d_in[17];
    const float* c4b2 = (const float*)d_in[18];
    const float* fc3w = (const float*)d_in[19];
    const float* fc3b = (const float*)d_in[20];
    const float* outw = (const float*)d_in[21];
    const float* outb = (const float*)d_in[22];
    float* out = (float*)d_out;

    char* w = (char*)d_ws;
    unsigned* agg = (unsigned*)w;  w += (size_t)N * 16 * 4;   // reused c1 then c2
    float* h1  = (float*)w;        w += (size_t)N * 16 * 4;
    float* h2  = (float*)w;        w += (size_t)N * 8 * 4;
    float* sqn = (float*)w;        w += (size_t)N * 4;
    int*   knn = (int*)w;          w += (size_t)N * 8 * 4;
    float* h3  = (float*)w;        w += (size_t)N * 8 * 4;
    float* h4  = (float*)w;        w += (size_t)N * 16 * 4;

    // encoder: c1 (6->16->16) on given edges
    init_enc_kernel<<<(N * 16) / 256, 256, 0, stream>>>(agg, N * 16);
    edge_conv_edges<3, 16, 16><<<E0 / 256, 256, 0, stream>>>(
        x, ei, ei + E0, c1w1, c1b1, c1w2, c1b2, agg, E0);
    decode_relu_kernel<<<(N * 16) / 256, 256, 0, stream>>>(agg, h1, N * 16);

    // encoder: c2 (32->8->8)
    init_enc_kernel<<<(N * 8) / 256, 256, 0, stream>>>(agg, N * 8);
    edge_conv_edges<16, 8, 8><<<E0 / 256, 256, 0, stream>>>(
        h1, ei, ei + E0, c2w1, c2b1, c2w2, c2b2, agg, E0);
    decode_h2_sq_kernel<<<N / 256, 256, 0, stream>>>(agg, h2, sqn, N);

    // dynamic graph: global kNN(k=8) via fp32 WMMA Gram + fused top-8
    knn_wmma_kernel<<<N / (16 * 8), 256, 0, stream>>>(h2, sqn, knn, N);

    // decoder: c3 (16->8->8), c4 (16->16->16) over kNN edges
    edge_conv_knn<8, 8, 8><<<N / 256, 256, 0, stream>>>(
        h2, knn, c3w1, c3b1, c3w2, c3b2, h3, N);
    edge_conv_knn<8, 16, 16><<<N / 256, 256, 0, stream>>>(
        h3, knn, c4w1, c4b1, c4w2, c4b2, h4, N);

    // per-graph max pool + fc3/out head
    head_kernel<<<G, 128, 0, stream>>>(h4, fc3w, fc3b, outw, outb, out);
}